// GNN_82111184765508
// MI455X (gfx1250) — compile-verified
//
#include <hip/hip_runtime.h>
#include <math.h>

// ---------------------------------------------------------------------------
// 2-layer GCN for MI455X (gfx1250, wave32).
//   layer1: relu( Dinv (A+I) Dinv (x @ W1) + b1 )        H = 64
//   layer2: log_softmax( Dinv (A+I) Dinv (h @ W2) + b2 ) C = 47 (padded to 48)
// GEMMs use V_WMMA_F32_16X16X4_F32 with one wave per 16-row strip computing
// ALL output columns (4 resp. 3 WMMA accumulators sharing one A fragment).
// Edge aggregation = vectorized fp32 global atomics (L2-resident buffers).
// ---------------------------------------------------------------------------

typedef __attribute__((ext_vector_type(2))) float v2f;
typedef __attribute__((ext_vector_type(8))) float v8f;

#define GCN_IN  256
#define GCN_H   64
#define GCN_C   47
#define GCN_CP  48   // padded C

#define WMMA_F32(a, b, c)                                                     \
  __builtin_amdgcn_wmma_f32_16x16x4_f32(false, (a), false, (b), (short)0, (c), \
                                        false, false)

// ---------------- degree / normalization -----------------------------------

__global__ void k_deg_init(float* __restrict__ deg, int n) {
  int g = blockIdx.x * blockDim.x + threadIdx.x;
  if (g < n) deg[g] = 1.0f;  // self-loop
}

__global__ void k_deg_accum(const int* __restrict__ dst, float* __restrict__ deg, int e) {
  int g = blockIdx.x * blockDim.x + threadIdx.x;
  if (g < e) atomicAdd(&deg[dst[g]], 1.0f);
}

__global__ void k_dinv(float* __restrict__ deg, int n) {
  int g = blockIdx.x * blockDim.x + threadIdx.x;
  if (g < n) deg[g] = rsqrtf(deg[g]);  // deg >= 1 always (self-loop)
}

// ---------------- GEMM1 (fused epilogue): [N,256] x [256,64] ----------------
// grid = N/16, block = 32: one wave computes a full 16x64 strip with 4 WMMA
// accumulators reusing one A fragment per k-step. Epilogue writes both
//   Hout = x@W1                     (gather source for the edge scatter)
//   Acc  = b1 + (x@W1)*dinv^2      (self-loop initialized accumulator)

__global__ __launch_bounds__(32) void k_gemm1_fused(
    const float* __restrict__ X, const float* __restrict__ W,
    const float* __restrict__ bias, const float* __restrict__ dinv,
    float* __restrict__ Hout, float* __restrict__ Acc) {
  const int mt   = blockIdx.x;
  const int lane = threadIdx.x;
  const int half = lane >> 4;   // selects K pair: kk = k + 2*half
  const int idx  = lane & 15;
  const int row  = mt * 16 + idx;

  const float* __restrict__ xrow = X + (size_t)row * GCN_IN;
  v8f acc[4] = {};
  for (int k = 0; k < GCN_IN; k += 4) {
    const int kk = k + 2 * half;
    v2f a;
    a.x = xrow[kk];
    a.y = xrow[kk + 1];
    const float* __restrict__ w0 = W + (size_t)kk * GCN_H + idx;        // K=kk
    const float* __restrict__ w1 = W + (size_t)(kk + 1) * GCN_H + idx;  // K=kk+1
#pragma unroll
    for (int nt = 0; nt < 4; ++nt) {
      v2f b;
      b.x = w0[nt * 16];
      b.y = w1[nt * 16];
      acc[nt] = WMMA_F32(a, b, acc[nt]);
    }
  }
  // C/D layout: VGPR r holds M = r + 8*half, N = nt*16 + idx
#pragma unroll
  for (int r = 0; r < 8; ++r) {
    const int m  = mt * 16 + r + 8 * half;
    const float di  = dinv[m];
    const float di2 = di * di;
#pragma unroll
    for (int nt = 0; nt < 4; ++nt) {
      const int    c = nt * 16 + idx;
      const float  h = acc[nt][r];
      Hout[(size_t)m * GCN_H + c] = h;
      Acc[(size_t)m * GCN_H + c]  = bias[c] + h * di2;
    }
  }
}

// ---------------- GEMM2: relu([N,64]) x [64,47->48] -------------------------
// grid = N/16, one wave per 16x48 strip (3 accumulators). ReLU folded into A
// loads; B col 47 is zeroed so the pad column stays exactly 0 and EXEC stays
// all-ones for WMMA. (No init fusion here: GEMM2 still READS the layer-1
// accumulator buffer, so the layer-2 init must wait for kernel completion.)

__global__ __launch_bounds__(32) void k_gemm2(const float* __restrict__ Hin,
                                              const float* __restrict__ W,
                                              float* __restrict__ Hout) {
  const int mt   = blockIdx.x;
  const int lane = threadIdx.x;
  const int half = lane >> 4;
  const int idx  = lane & 15;
  const int row  = mt * 16 + idx;

  const float* __restrict__ hrow = Hin + (size_t)row * GCN_H;
  v8f acc[3] = {};
  for (int k = 0; k < GCN_H; k += 4) {
    const int kk = k + 2 * half;
    v2f a;
    a.x = fmaxf(hrow[kk], 0.0f);       // ReLU from layer 1
    a.y = fmaxf(hrow[kk + 1], 0.0f);
#pragma unroll
    for (int nt = 0; nt < 3; ++nt) {
      const int col = nt * 16 + idx;
      const bool ok = (col < GCN_C);   // only tile 2 / idx 15 is padding
      v2f b;
      b.x = ok ? W[(size_t)kk * GCN_C + col] : 0.0f;
      b.y = ok ? W[(size_t)(kk + 1) * GCN_C + col] : 0.0f;
      acc[nt] = WMMA_F32(a, b, acc[nt]);
    }
  }
#pragma unroll
  for (int r = 0; r < 8; ++r) {
    const int m = mt * 16 + r + 8 * half;
#pragma unroll
    for (int nt = 0; nt < 3; ++nt)
      Hout[(size_t)m * GCN_CP + nt * 16 + idx] = acc[nt][r];
  }
}

// ---------------- layer-2 self-loop init ------------------------------------

__global__ void k_agg_init48(const float* __restrict__ h, const float* __restrict__ dinv,
                             const float* __restrict__ bias, float* __restrict__ out,
                             unsigned total) {
  unsigned g = blockIdx.x * blockDim.x + threadIdx.x;
  if (g >= total) return;
  unsigned i = g / GCN_CP, f = g % GCN_CP;
  float di = dinv[i];
  float b  = (f < GCN_C) ? bias[f] : 0.0f;  // pad col of h is already 0
  out[g] = b + h[g] * di * di;
}

// ---------------- edge scatter: 1 thread = 1 edge x 2 features --------------
// b64 gather from h[src], two fp32 atomics into out[dst]; all buffers are
// L2-resident (25.6/19.2 MB << 192 MB), so this is an L2-atomic phase.
// For F=64 a whole wave owns one edge -> src/dst/dinv loads are wave-uniform.

template <int F>
__global__ void k_edge_scatter(const int* __restrict__ src, const int* __restrict__ dst,
                               const float* __restrict__ h, const float* __restrict__ dinv,
                               float* __restrict__ out, unsigned total) {
  unsigned g = blockIdx.x * blockDim.x + threadIdx.x;
  if (g >= total) return;
  constexpr unsigned FP = F / 2;
  unsigned e = g / FP;
  unsigned f = (g % FP) * 2;
  int s = src[e], d = dst[e];
  float norm = dinv[s] * dinv[d];
  const float2 hv = *(const float2*)(h + (size_t)s * F + f);
  float* o = out + (size_t)d * F + f;
  atomicAdd(o,     hv.x * norm);
  atomicAdd(o + 1, hv.y * norm);
}

// ---------------- log-softmax over 47 cols, one wave32 per row --------------

__global__ void k_logsoftmax(const float* __restrict__ in /*stride 48*/,
                             float* __restrict__ out /*stride 47*/) {
  const int wave = threadIdx.x >> 5;
  const int lane = threadIdx.x & 31;
  const int row  = blockIdx.x * 8 + wave;
  const float* __restrict__ r = in + (size_t)row * GCN_CP;

  float v1 = r[lane];
  float v2 = (lane < GCN_C - 32) ? r[lane + 32] : -INFINITY;

  float m = fmaxf(v1, v2);
#pragma unroll
  for (int off = 16; off; off >>= 1) m = fmaxf(m, __shfl_xor(m, off, 32));

  float s = expf(v1 - m) + ((lane < GCN_C - 32) ? expf(v2 - m) : 0.0f);
#pragma unroll
  for (int off = 16; off; off >>= 1) s += __shfl_xor(s, off, 32);

  const float ls = logf(s);
  out[(size_t)row * GCN_C + lane] = v1 - m - ls;
  if (lane < GCN_C - 32) out[(size_t)row * GCN_C + lane + 32] = v2 - m - ls;
}

// ---------------------------------------------------------------------------

static inline unsigned cdiv(unsigned a, unsigned b) { return (a + b - 1) / b; }

extern "C" void kernel_launch(void* const* d_in, const int* in_sizes, int n_in,
                              void* d_out, int out_size, void* d_ws, size_t ws_size,
                              hipStream_t stream) {
  const float* x   = (const float*)d_in[0];   // [N, 256]
  const float* W1  = (const float*)d_in[1];   // [256, 64]
  const float* b1  = (const float*)d_in[2];   // [64]
  const float* W2  = (const float*)d_in[3];   // [64, 47]
  const float* b2  = (const float*)d_in[4];   // [47]
  const int*   ei  = (const int*)d_in[5];     // [2, E]
  float*       out = (float*)d_out;           // [N, 47]

  const int n = in_sizes[0] / GCN_IN;         // 100000 (divisible by 16)
  const int e = in_sizes[5] / 2;              // 1600000
  const int* src = ei;
  const int* dst = ei + e;

  // workspace (fp32): dinv[N] | bufA[N*64] | bufB[N*64]   (~52 MB total)
  // bufA: h1, reused as h2 (stride 48);  bufB: agg1, reused as agg2 (stride 48)
  float* dinv = (float*)d_ws;
  float* bufA = dinv + ((n + 255) & ~255);
  float* bufB = bufA + (size_t)n * GCN_H;

  const int B = 256;

  // 1) symmetric normalization coefficients
  k_deg_init<<<cdiv(n, B), B, 0, stream>>>(dinv, n);
  k_deg_accum<<<cdiv(e, B), B, 0, stream>>>(dst, dinv, e);
  k_dinv<<<cdiv(n, B), B, 0, stream>>>(dinv, n);

  // 2) layer 1: WMMA GEMM with fused self-loop init, then edge scatter
  k_gemm1_fused<<<n / 16, 32, 0, stream>>>(x, W1, b1, dinv, bufA, bufB);
  {
    unsigned etot = (unsigned)e * (GCN_H / 2);
    k_edge_scatter<GCN_H><<<cdiv(etot, B), B, 0, stream>>>(src, dst, bufA, dinv, bufB, etot);
  }

  // 3) layer 2: WMMA GEMM (ReLU folded in), init, edge scatter
  k_gemm2<<<n / 16, 32, 0, stream>>>(bufB, W2, bufA);
  {
    unsigned tot = (unsigned)n * GCN_CP;
    k_agg_init48<<<cdiv(tot, B), B, 0, stream>>>(bufA, dinv, b2, bufB, tot);
    unsigned etot = (unsigned)e * (GCN_CP / 2);
    k_edge_scatter<GCN_CP><<<cdiv(etot, B), B, 0, stream>>>(src, dst, bufA, dinv, bufB, etot);
  }

  // 4) row-wise log-softmax -> final output [N, 47]
  k_logsoftmax<<<n / 8, B, 0, stream>>>(bufB, out);
}